// VectorQuantizer_72945724555846
// MI455X (gfx1250) — compile-verified
//
#include <hip/hip_runtime.h>
#include <hip/hip_bf16.h>
#include <float.h>
#include <math.h>

typedef __attribute__((ext_vector_type(2))) float v2f;
typedef __attribute__((ext_vector_type(8))) float v8f;

#define VOCAB    1024
#define EMBED    64
#define NVEC     65536      // 64*32*32 latent vectors
#define OUTELEMS 4194304    // 64*64*32*32

// workspace layout (byte offsets)
#define WS_ESQ   0                       // float[1024]
#define WS_IDX   4096                    // int[65536]
#define WS_CNT   266240                  // uint[1024]
#define WS_PART  270336                  // float[16384]
#define NPART    16384

#define CHUNK      128
#define LDS_STRIDE 68   // 64 + 4 pad -> bank stride 4, conflict-free code-row gather

// ---------------- prep: per-code squared norms ----------------
__global__ __launch_bounds__(256) void vq_prep_esq(const float* __restrict__ emb,
                                                   float* __restrict__ esq) {
  int v = blockIdx.x * 256 + threadIdx.x;
  if (v < VOCAB) {
    const float4* r = (const float4*)(emb + (size_t)v * EMBED);
    float s = 0.f;
#pragma unroll
    for (int i = 0; i < 16; ++i) {
      float4 x = r[i];
      s += x.x * x.x + x.y * x.y + x.z * x.z + x.w * x.w;
    }
    esq[v] = s;
  }
}

__global__ void vq_zero_counts(unsigned* __restrict__ cnt) {
  cnt[threadIdx.x] = 0u;   // launched with 1024 threads
}

// ---------------- main: WMMA distance + argmin ----------------
// Each wave handles 16 consecutive rows (16 consecutive w within one (b,h)).
// Codebook streamed through LDS in 128-code chunks; two code tiles processed
// concurrently -> two independent WMMA D->C accumulation chains for ILP.
__global__ __launch_bounds__(256)
void vq_argmin(const float* __restrict__ z, const float* __restrict__ emb,
               const float* __restrict__ esq, int* __restrict__ idx_out,
               float* __restrict__ idx_f_out, unsigned* __restrict__ counts) {
  __shared__ float lds_e[CHUNK * LDS_STRIDE];
  __shared__ float lds_q[CHUNK];

  const int tid  = threadIdx.x;
  const int wave = tid >> 5;
  const int lid  = tid & 31;
  const int col  = lid & 15;
  const int hi   = lid >> 4;

  const int n0  = (blockIdx.x * 8 + wave) * 16;
  const int row = n0 + col;
  const int b   = row >> 10;
  const int hw  = row & 1023;               // h*32 + w
  const float* zrow = z + (size_t)b * 65536 + hw;  // element (row, d) at zrow + d*1024

  // A tile: a[k] holds z[row, 4k+2*hi] , z[row, 4k+2*hi+1]  (16x4 f32 WMMA layout)
  v2f a[16];
#pragma unroll
  for (int k = 0; k < 16; ++k) {
    int d0 = 4 * k + 2 * hi;
    a[k].x = zrow[(size_t)d0 * 1024];
    a[k].y = zrow[(size_t)(d0 + 1) * 1024];
  }

  float bs[8];
  int   bi[8];
#pragma unroll
  for (int j = 0; j < 8; ++j) { bs[j] = FLT_MAX; bi[j] = 0; }

  for (int ch = 0; ch < VOCAB / CHUNK; ++ch) {
    __syncthreads();
    // stage 128 code rows (coalesced float4), padded stride in LDS
    for (int i = tid; i < CHUNK * EMBED / 4; i += 256) {
      int g = i * 4;
      int r = g >> 6;
      int c = g & 63;
      float4 v4 = *(const float4*)(emb + (size_t)ch * CHUNK * EMBED + g);
      *(float4*)(&lds_e[r * LDS_STRIDE + c]) = v4;
    }
    if (tid < CHUNK) lds_q[tid] = esq[ch * CHUNK + tid];
    __syncthreads();

#pragma unroll 1
    for (int tt = 0; tt < CHUNK / 16; tt += 2) {
      const int lc0 = tt * 16 + col;          // this lane's code column, tile 0
      const int lc1 = lc0 + 16;               // tile 1
      const float* bp0 = &lds_e[lc0 * LDS_STRIDE + 2 * hi];
      const float* bp1 = &lds_e[lc1 * LDS_STRIDE + 2 * hi];
      v8f c8a = {0.f, 0.f, 0.f, 0.f, 0.f, 0.f, 0.f, 0.f};
      v8f c8b = {0.f, 0.f, 0.f, 0.f, 0.f, 0.f, 0.f, 0.f};
#pragma unroll
      for (int k = 0; k < 16; ++k) {   // K = 64 in 16 steps of 4, two chains
        v2f b0, b1;
        b0.x = bp0[4 * k];
        b0.y = bp0[4 * k + 1];
        b1.x = bp1[4 * k];
        b1.y = bp1[4 * k + 1];
        c8a = __builtin_amdgcn_wmma_f32_16x16x4_f32(false, a[k], false, b0,
                                                    (short)0, c8a, false, false);
        c8b = __builtin_amdgcn_wmma_f32_16x16x4_f32(false, a[k], false, b1,
                                                    (short)0, c8b, false, false);
      }
      const float es0 = lds_q[lc0];
      const float es1 = lds_q[lc1];
      const int vc0 = ch * CHUNK + tt * 16 + col;
      const int vc1 = vc0 + 16;
#pragma unroll
      for (int j = 0; j < 8; ++j) {
        float s0 = fmaf(-2.f, c8a[j], es0);   // ||e||^2 - 2 z.e  (||z||^2 drops out)
        if (s0 < bs[j]) { bs[j] = s0; bi[j] = vc0; }  // strict < keeps first min
        float s1 = fmaf(-2.f, c8b[j], es1);
        if (s1 < bs[j]) { bs[j] = s1; bi[j] = vc1; }
      }
    }
  }

  // butterfly argmin across the 16 lanes sharing each row (tie -> smaller index)
#pragma unroll
  for (int j = 0; j < 8; ++j) {
#pragma unroll
    for (int off = 1; off < 16; off <<= 1) {
      float os = __shfl_xor(bs[j], off, 32);
      int   ov = __shfl_xor(bi[j], off, 32);
      if (os < bs[j] || (os == bs[j] && ov < bi[j])) { bs[j] = os; bi[j] = ov; }
    }
  }

  if (col == 0) {   // lane 0 -> rows n0..n0+7 ; lane 16 -> rows n0+8..n0+15
    int base = n0 + hi * 8;
#pragma unroll
    for (int j = 0; j < 8; ++j) {
      int v = bi[j];
      idx_out[base + j]   = v;
      idx_f_out[base + j] = (float)v;
      atomicAdd(&counts[v], 1u);
    }
  }
}

// ---------------- gather z_q + deterministic loss partials ----------------
__global__ __launch_bounds__(256)
void vq_gather(const float* __restrict__ z, const float* __restrict__ emb,
               const int* __restrict__ idx, float* __restrict__ zq_out,
               float* __restrict__ partial) {
  __shared__ float red[256];
  int o = blockIdx.x * 256 + threadIdx.x;
  int w = o & 31, h = (o >> 5) & 31, d = (o >> 10) & 63, b = o >> 16;
  int n = (b << 10) | (h << 5) | w;
  int v = idx[n];
  float q = emb[(size_t)v * EMBED + d];
  float e = z[o];
  zq_out[o] = q;               // forward value of straight-through output == z_q
  float diff = q - e;
  red[threadIdx.x] = diff * diff;
  __syncthreads();
  for (int s = 128; s > 0; s >>= 1) {
    if (threadIdx.x < s) red[threadIdx.x] += red[threadIdx.x + s];
    __syncthreads();
  }
  if (threadIdx.x == 0) partial[blockIdx.x] = red[0];
}

// ---------------- finalize: losses + perplexity ----------------
__global__ __launch_bounds__(1024)
void vq_finalize(const float* __restrict__ partial, const unsigned* __restrict__ counts,
                 float* __restrict__ out_scalars) {
  __shared__ float red[1024];
  int t = threadIdx.x;
  float s = 0.f;
  for (int i = t; i < NPART; i += 1024) s += partial[i];
  red[t] = s;
  __syncthreads();
  for (int k = 512; k > 0; k >>= 1) { if (t < k) red[t] += red[t + k]; __syncthreads(); }
  float sumsq = red[0];
  __syncthreads();
  float p = (float)counts[t] / (float)NVEC;
  red[t] = p * logf(p + 1e-10f);
  __syncthreads();
  for (int k = 512; k > 0; k >>= 1) { if (t < k) red[t] += red[t + k]; __syncthreads(); }
  if (t == 0) {
    float ql = sumsq / (float)OUTELEMS;
    out_scalars[0] = ql;          // quantization_loss
    out_scalars[1] = ql * 0.25f;  // commitment_loss (same forward value * beta)
    out_scalars[2] = expf(-red[0]);
  }
}

extern "C" void kernel_launch(void* const* d_in, const int* in_sizes, int n_in,
                              void* d_out, int out_size, void* d_ws, size_t ws_size,
                              hipStream_t stream) {
  const float* z   = (const float*)d_in[0];   // [64,64,32,32]
  const float* emb = (const float*)d_in[1];   // [1024,64]
  float* out = (float*)d_out;
  char*  ws  = (char*)d_ws;

  float*    esq  = (float*)(ws + WS_ESQ);
  int*      idx  = (int*)(ws + WS_IDX);
  unsigned* cnt  = (unsigned*)(ws + WS_CNT);
  float*    part = (float*)(ws + WS_PART);

  float* zq_out  = out;                       // [64,64,32,32]
  float* idxf    = out + OUTELEMS;            // [64,32,32] as float
  float* scalars = out + OUTELEMS + NVEC;     // 3 scalars

  vq_prep_esq   <<<4, 256, 0, stream>>>(emb, esq);
  vq_zero_counts<<<1, 1024, 0, stream>>>(cnt);
  vq_argmin     <<<NVEC / 128, 256, 0, stream>>>(z, emb, esq, idx, idxf, cnt);
  vq_gather     <<<OUTELEMS / 256, 256, 0, stream>>>(z, emb, idx, zq_out, part);
  vq_finalize   <<<1, 1024, 0, stream>>>(part, cnt, scalars);
}